// SepGPDLayer_39419209843204
// MI455X (gfx1250) — compile-verified
//
#include <hip/hip_runtime.h>
#include <math.h>

// SepGPD layer: per point p in (B,H,W)=(4,64,64):
//   m[p][c]   = x[b][c][h][w],                    c in [0,16)
//   d[p][c]   = smin*(1-sig) + smax*sig,          c in [16,32)
//   a[p][k]   = pi * tanh(x[b][32+k][h][w]),      k in [0,120)
//   R = Givens(0,1)...Givens(14,15) column-update chain on I16
//   S = R^T diag(d) R  (symmetrized)
// One wave32 handles one point. R lives in WMMA C/D layout; S via
// 4x v_wmma_f32_16x16x4_f32 of G^T G with G = sqrt(d)-scaled R.

typedef float v2f __attribute__((ext_vector_type(2)));
typedef float v8f __attribute__((ext_vector_type(8)));

#define DIM   16
#define NANG  120           // DIM*(DIM-1)/2
#define NCH   152           // 2*DIM + NANG
#define HW    4096          // 64*64
#define NPTS  16384         // 4*64*64
#define WPB   8             // waves per block
// per-wave LDS (floats): [0,240) cos/sin pairs, [240,256) sqrt(d), [256,512) 16x16 tile
#define WSLOT 512

__global__ __launch_bounds__(256) void sepgpd_kernel(const float* __restrict__ x,
                                                     float* __restrict__ out_m,
                                                     float* __restrict__ out_S) {
    const int lane = threadIdx.x & 31;
    const int wid  = threadIdx.x >> 5;
    const int p    = blockIdx.x * WPB + wid;     // point id
    const int b    = p >> 12;                    // / 4096
    const int sp   = p & 4095;                   // h*64 + w
    const float* xp = x + (size_t)b * NCH * HW + sp;

    __shared__ float lds[WPB * WSLOT];
    float* wlds   = lds + wid * WSLOT;
    float* cs_tab = wlds;            // 120 (cos,sin) pairs
    float* d_tab  = wlds + 240;      // 16 sqrt(d)
    float* tile   = wlds + 256;      // 16x16 scratch tile

    // ---- stage 1: load channels, transcendentals, fill LDS tables ----
    {
        float v = xp[(size_t)lane * HW];         // channel = lane
        if (lane < DIM) {
            out_m[(size_t)p * DIM + lane] = v;   // mean pass-through
        } else {
            float sg = 1.0f / (1.0f + __expf(-v));
            float dv = 0.001f * (1.0f - sg) + 1000.0f * sg;
            d_tab[lane - DIM] = sqrtf(dv);
        }
    }
#pragma unroll
    for (int q = 0; q < 4; ++q) {
        int k = lane + 32 * q;
        if (k < NANG) {
            float v = xp[(size_t)(2 * DIM + k) * HW];
            float a = 3.14159265358979323846f * tanhf(v);
            cs_tab[2 * k]     = __cosf(a);
            cs_tab[2 * k + 1] = __sinf(a);
        }
    }
    __syncthreads();

    // ---- stage 2: Givens column-update chain on R (C/D layout) ----
    // lane l, reg v  <->  R[v + 8*(l>>4)][l & 15]
    const int col  = lane & 15;
    const int half = lane >> 4;
    float r[8];
#pragma unroll
    for (int v = 0; v < 8; ++v) r[v] = ((v + 8 * half) == col) ? 1.0f : 0.0f;

    int k = 0;
#pragma unroll 1
    for (int i = 0; i < DIM - 1; ++i) {
#pragma unroll 1
        for (int j = i + 1; j < DIM; ++j, ++k) {
            float c = cs_tab[2 * k];
            float s = cs_tab[2 * k + 1];
            bool is_i = (col == i);
            bool is_j = (col == j);
            int delta = j - i;
            int partner = lane + (is_i ? delta : 0) - (is_j ? delta : 0);
            float ssel = is_j ? -s : s;          // col_i' = c*ci + s*cj ; col_j' = c*cj - s*ci
            bool act = is_i || is_j;
#pragma unroll
            for (int v = 0; v < 8; ++v) {
                float other = __shfl(r[v], partner, 32);   // ds_bpermute, all lanes read
                float t = fmaf(ssel, other, c * r[v]);
                r[v] = act ? t : r[v];
            }
        }
    }

    // ---- stage 3: G = sqrt(d)-row-scaled R -> LDS tile ----
#pragma unroll
    for (int v = 0; v < 8; ++v) {
        int row = v + 8 * half;
        tile[row * 16 + col] = r[v] * d_tab[row];
    }
    __syncthreads();

    // ---- stage 4: S = G^T G via 4x WMMA f32 16x16x4 ----
    // 32-bit 16x4 A layout: lane l, reg t -> (M = l&15, K = t + 2*(l>>4)).
    // For A = G^T chunk and B = G chunk these per-lane register contents
    // coincide, so one LDS fetch feeds both operands.
    v8f acc = {};
#pragma unroll
    for (int kc = 0; kc < 4; ++kc) {
        v2f ab;
        ab.x = tile[(kc * 4 + 2 * half + 0) * 16 + col];
        ab.y = tile[(kc * 4 + 2 * half + 1) * 16 + col];
        acc = __builtin_amdgcn_wmma_f32_16x16x4_f32(
            /*neg_a=*/false, ab, /*neg_b=*/false, ab,
            /*c_mod=*/(short)0, acc, /*reuse_a=*/false, /*reuse_b=*/false);
    }
    __syncthreads();

    // ---- stage 5: symmetrize 0.5*(S + S^T) via LDS transpose ----
#pragma unroll
    for (int v = 0; v < 8; ++v)
        tile[(v + 8 * half) * 16 + col] = acc[v];
    __syncthreads();

    float* Sp = out_S + (size_t)p * 256;
#pragma unroll
    for (int v = 0; v < 8; ++v) {
        float tr = tile[col * 16 + (v + 8 * half)];
        Sp[(v + 8 * half) * 16 + col] = 0.5f * (acc[v] + tr);
    }
}

extern "C" void kernel_launch(void* const* d_in, const int* in_sizes, int n_in,
                              void* d_out, int out_size, void* d_ws, size_t ws_size,
                              hipStream_t stream) {
    (void)in_sizes; (void)n_in; (void)d_ws; (void)ws_size; (void)out_size;
    const float* x = (const float*)d_in[0];
    float* out_m = (float*)d_out;                       // (4,64,64,16)   = 262144 f32
    float* out_S = out_m + (size_t)NPTS * DIM;          // (4,64,64,16,16)= 4194304 f32
    sepgpd_kernel<<<dim3(NPTS / WPB), dim3(256), 0, stream>>>(x, out_m, out_S);
}